// InternalGraphConvolutionLayer_17111149707559
// MI455X (gfx1250) — compile-verified
//
#include <hip/hip_runtime.h>
#include <hip/hip_bf16.h>

typedef __attribute__((ext_vector_type(2))) float v2f;
typedef __attribute__((ext_vector_type(8))) float v8f;

// ---------------------------------------------------------------------------
// Kernel 0: zero the atomic accumulation regions (agg[N*3] ++ colsum[4]).
// ---------------------------------------------------------------------------
__global__ __launch_bounds__(256) void gc_zero_kernel(float* __restrict__ p, int n) {
    int i = blockIdx.x * 256 + threadIdx.x;
    if (i < n) p[i] = 0.0f;
}

// ---------------------------------------------------------------------------
// Kernel 1: xw = x@W, xm = x@M via V_WMMA_F32_16X16X4_F32.
// 8 waves per block, one 16-node tile per wave32. FULL TILES ONLY — no
// bounds checks anywhere on this path (tail handled by gc_tail_kernel).
//
// A layout (32-bit A 16x4, ISA 7.12.2): lane L in [0,15] -> row M=L,
//   v0 = K=0, v1 = K=1; lane L+16 -> row M=L, v0 = K=2, v1 = K=3.
// B layout (4x16): VGPR j -> row K=j (lanes 0-15), K=j+2 (lanes 16-31),
//   col N = lane%16.
// D layout (32-bit 16x16): VGPR v -> row M=v (lanes 0-15) / M=v+8
//   (lanes 16-31), col N = lane%16.
// ---------------------------------------------------------------------------
__global__ __launch_bounds__(256) void gc_xwxm_wmma_kernel(
    const float* __restrict__ x,
    const float* __restrict__ W,
    const float* __restrict__ M,
    float* __restrict__ xw,
    float* __restrict__ xm) {
    const int lane = threadIdx.x & 31;                    // lane in wave32
    const int wave = threadIdx.x >> 5;                    // wave in block
    const int tile = blockIdx.x * 8 + wave;               // 16-node tile id
    const int base = tile * 16;                           // first node
    const int m    = lane & 15;                           // tile row / col N
    const bool hi  = lane >= 16;

    // ---- A operand: x rows, K padded 3->4 with zero (branchless selects)
    const int node = base + m;
    float a0 = hi ? x[node * 3 + 2] : x[node * 3 + 0];
    float a1 = hi ? 0.0f            : x[node * 3 + 1];
    v2f a;  a.x = a0;  a.y = a1;

    // ---- B operands: 3x3 W / M zero-padded into 4x16, branchless.
    // kbase = hi ? 2 : 0;  k0 = kbase (<3 always), k1 = kbase+1 (<3 iff !hi)
    const int n   = m;                                    // output feature
    const bool c0 = (n < 3);
    const bool c1 = (n < 3) && !hi;
    const int kbase = hi ? 2 : 0;
    const int i0 = c0 ? (kbase * 3 + n) : 0;              // clamped, in-bounds
    const int i1 = c1 ? ((kbase + 1) * 3 + n) : 0;
    float w0 = W[i0], w1 = W[i1];
    float m0 = M[i0], m1 = M[i1];
    v2f bw, bm;
    bw.x = c0 ? w0 : 0.0f;  bw.y = c1 ? w1 : 0.0f;
    bm.x = c0 ? m0 : 0.0f;  bm.y = c1 ? m1 : 0.0f;

    v8f zero = {};
    // D = A x B + 0  (neg_a, A, neg_b, B, c_mod, C, reuse_a, reuse_b)
    v8f dw = __builtin_amdgcn_wmma_f32_16x16x4_f32(
        false, a, false, bw, (short)0, zero, false, false);
    v8f dm = __builtin_amdgcn_wmma_f32_16x16x4_f32(
        false, a, false, bm, (short)0, zero, false, false);

    // ---- Scatter D: lanes with n<3 own feature n for rows v (+8 if hi).
    // One divergent region, straight-line unrolled stores inside.
    if (n < 3) {
        const int rbase = (base + (hi ? 8 : 0)) * 3 + n;
#pragma unroll
        for (int v = 0; v < 8; ++v) {
            xw[rbase + v * 3] = dw[v];
            xm[rbase + v * 3] = dm[v];
        }
    }
}

// ---------------------------------------------------------------------------
// Kernel 1b: scalar tail for nodes not covered by full 16-node tiles.
// (Launched for N=500000: the last 32 nodes, as 31250 tiles % 8 = 2.)
// ---------------------------------------------------------------------------
__global__ __launch_bounds__(64) void gc_tail_kernel(
    const float* __restrict__ x,
    const float* __restrict__ W,
    const float* __restrict__ M,
    float* __restrict__ xw,
    float* __restrict__ xm,
    int start, int n_nodes) {
    int node = start + blockIdx.x * 64 + threadIdx.x;
    if (node >= n_nodes) return;
    float x0 = x[node * 3 + 0], x1 = x[node * 3 + 1], x2 = x[node * 3 + 2];
#pragma unroll
    for (int f = 0; f < 3; ++f) {
        xw[node * 3 + f] = fmaf(x0, W[f], fmaf(x1, W[3 + f], x2 * W[6 + f]));
        xm[node * 3 + f] = fmaf(x0, M[f], fmaf(x1, M[3 + f], x2 * M[6 + f]));
    }
}

// ---------------------------------------------------------------------------
// Kernel 2: edge scatter. src/dst streamed once -> non-temporal loads so the
// 192MB L2 keeps xm (6MB) and agg (6MB) resident. Non-returning fp32 atomics.
// ---------------------------------------------------------------------------
__global__ __launch_bounds__(256) void gc_edge_kernel(
    const int* __restrict__ edge_src,
    const int* __restrict__ edge_dst,
    const float* __restrict__ xm,
    float* __restrict__ agg,
    int n_edges) {
    int e = blockIdx.x * 256 + threadIdx.x;
    if (e >= n_edges) return;
    int s = __builtin_nontemporal_load(edge_src + e);
    int d = __builtin_nontemporal_load(edge_dst + e);
    float v0 = xm[s * 3 + 0];
    float v1 = xm[s * 3 + 1];
    float v2 = xm[s * 3 + 2];
    unsafeAtomicAdd(&agg[d * 3 + 0], v0);
    unsafeAtomicAdd(&agg[d * 3 + 1], v1);
    unsafeAtomicAdd(&agg[d * 3 + 2], v2);
}

// ---------------------------------------------------------------------------
// Kernel 3: h = relu(xw + agg); LDS block reduction of the 3 column sums,
// then 3 atomic adds per block into colsum.
// ---------------------------------------------------------------------------
__global__ __launch_bounds__(256) void gc_relu_sum_kernel(
    const float* __restrict__ xw,
    const float* __restrict__ agg,
    float* __restrict__ colsum,
    int n_nodes) {
    __shared__ float s0[256], s1[256], s2[256];
    int i = blockIdx.x * 256 + threadIdx.x;
    float h0 = 0.0f, h1 = 0.0f, h2 = 0.0f;
    if (i < n_nodes) {
        h0 = fmaxf(xw[i * 3 + 0] + agg[i * 3 + 0], 0.0f);
        h1 = fmaxf(xw[i * 3 + 1] + agg[i * 3 + 1], 0.0f);
        h2 = fmaxf(xw[i * 3 + 2] + agg[i * 3 + 2], 0.0f);
    }
    int t = threadIdx.x;
    s0[t] = h0; s1[t] = h1; s2[t] = h2;
    __syncthreads();
#pragma unroll
    for (int off = 128; off > 0; off >>= 1) {
        if (t < off) {
            s0[t] += s0[t + off];
            s1[t] += s1[t + off];
            s2[t] += s2[t + off];
        }
        __syncthreads();
    }
    if (t == 0) {
        unsafeAtomicAdd(&colsum[0], s0[0]);
        unsafeAtomicAdd(&colsum[1], s1[0]);
        unsafeAtomicAdd(&colsum[2], s2[0]);
    }
}

// ---------------------------------------------------------------------------
// Kernel 4: softmax over the 3 column sums -> d_out[0..2].
// ---------------------------------------------------------------------------
__global__ void gc_softmax_kernel(const float* __restrict__ colsum,
                                  float* __restrict__ out) {
    if (threadIdx.x == 0 && blockIdx.x == 0) {
        float a = colsum[0], b = colsum[1], c = colsum[2];
        float mx = fmaxf(a, fmaxf(b, c));
        float e0 = __expf(a - mx);
        float e1 = __expf(b - mx);
        float e2 = __expf(c - mx);
        float inv = 1.0f / (e0 + e1 + e2);
        out[0] = e0 * inv;
        out[1] = e1 * inv;
        out[2] = e2 * inv;
    }
}

// ---------------------------------------------------------------------------
// Launch: inputs in setup_inputs() order: x[N*3], W[9], M[9], edge_src[E],
// edge_dst[E]. Workspace layout: xw[N*3] | xm[N*3] | agg[N*3] | colsum[4].
// ---------------------------------------------------------------------------
extern "C" void kernel_launch(void* const* d_in, const int* in_sizes, int n_in,
                              void* d_out, int out_size, void* d_ws, size_t ws_size,
                              hipStream_t stream) {
    const float* x  = (const float*)d_in[0];
    const float* W  = (const float*)d_in[1];
    const float* M  = (const float*)d_in[2];
    const int* esrc = (const int*)d_in[3];
    const int* edst = (const int*)d_in[4];

    const int n_nodes = in_sizes[0] / 3;
    const int n_edges = in_sizes[3];

    float* ws     = (float*)d_ws;
    float* xw     = ws;
    float* xm     = ws + (size_t)n_nodes * 3;
    float* agg    = ws + (size_t)n_nodes * 3 * 2;
    float* colsum = ws + (size_t)n_nodes * 3 * 3;

    // 1) zero agg + colsum (contiguous: n_nodes*3 + 4 floats)
    {
        int nz = n_nodes * 3 + 4;
        gc_zero_kernel<<<(nz + 255) / 256, 256, 0, stream>>>(agg, nz);
    }
    // 2) WMMA: xw = x@W, xm = x@M (full 16-node tiles, 8 tiles per block)
    {
        int full_tiles = n_nodes / 16;
        if (full_tiles > 0) {
            int blocks = (full_tiles + 7) / 8;
            // guard: last block's waves must all map to valid tiles
            if (full_tiles % 8 == 0) {
                gc_xwxm_wmma_kernel<<<blocks, 256, 0, stream>>>(x, W, M, xw, xm);
            } else {
                // launch whole blocks for the aligned part, scalar tail after
                int aligned_tiles = (full_tiles / 8) * 8;
                if (aligned_tiles > 0)
                    gc_xwxm_wmma_kernel<<<aligned_tiles / 8, 256, 0, stream>>>(x, W, M, xw, xm);
                int start = aligned_tiles * 16;
                int rem = n_nodes - start;
                gc_tail_kernel<<<(rem + 63) / 64, 64, 0, stream>>>(x, W, M, xw, xm, start, n_nodes);
                rem = 0;
            }
        }
        int tail_start = (n_nodes / 16) * 16;
        if (tail_start < n_nodes && (n_nodes / 16) % 8 == 0) {
            int rem = n_nodes - tail_start;
            gc_tail_kernel<<<(rem + 63) / 64, 64, 0, stream>>>(x, W, M, xw, xm, tail_start, n_nodes);
        }
    }
    // 3) edge scatter-add into agg
    gc_edge_kernel<<<(n_edges + 255) / 256, 256, 0, stream>>>(esrc, edst, xm, agg, n_edges);
    // 4) relu + column reduction
    gc_relu_sum_kernel<<<(n_nodes + 255) / 256, 256, 0, stream>>>(xw, agg, colsum, n_nodes);
    // 5) softmax of 3 values
    gc_softmax_kernel<<<1, 32, 0, stream>>>(colsum, (float*)d_out);
}